// PointConv_36816459661390
// MI455X (gfx1250) — compile-verified
//
#include <hip/hip_runtime.h>

// Problem constants (match reference)
#define BB    16
#define NN    16384
#define SS    4096
#define KK    16
#define DOUT  128
#define KTOT  1072   // 67 * 16

typedef float v2f __attribute__((ext_vector_type(2)));
typedef float v8f __attribute__((ext_vector_type(8)));

__device__ __forceinline__ float leaky02(float x) { return x > 0.f ? x : 0.2f * x; }

__global__ __launch_bounds__(256)
void pointconv_fused(const float* __restrict__ xyz,
                     const float* __restrict__ points,
                     const float* __restrict__ new_xyz,
                     const int*   __restrict__ nn_idx,
                     const float* __restrict__ w0, const float* __restrict__ b0,
                     const float* __restrict__ g0, const float* __restrict__ be0,
                     const float* __restrict__ w1, const float* __restrict__ b1,
                     const float* __restrict__ g1, const float* __restrict__ be1,
                     const float* __restrict__ w2, const float* __restrict__ b2,
                     const float* __restrict__ g2, const float* __restrict__ be2,
                     const float* __restrict__ wl, const float* __restrict__ bl,
                     const float* __restrict__ gl, const float* __restrict__ bel,
                     float* __restrict__ out)
{
    __shared__ int   lds_idx[256];          // [16 s][16 k]
    __shared__ float lds_gxn[16 * 3 * 16];  // [s][3][k]
    __shared__ float lds_w[16 * 256];       // [s][k][m]  weightnet output
    __shared__ float lds_feat[16 * 256];    // [s][16 c][16 k] per channel-tile
    __shared__ float lds_agg[16 * 256];     // [s][16 c][16 m]
    __shared__ float lds_rsum[16 * 8];
    __shared__ float lds_rsq[16 * 8];
    __shared__ float lds_mean[16];
    __shared__ float lds_rstd[16];

    const int t    = threadIdx.x;
    const int wave = t >> 5;
    const int lane = t & 31;
    const int lh   = lane >> 4;   // which 16-lane half
    const int ll   = lane & 15;

    const int blk   = blockIdx.x;
    const int b     = blk >> 8;          // / (S/16) = 256 tiles per batch
    const int stile = blk & 255;
    const int s0    = stile * 16;

    // ---------- Phase A: WeightNet MLP, one (s,k) per thread ----------
    {
        const int sl = t >> 4;           // local s 0..15
        const int k  = t & 15;
        const int s  = s0 + sl;
        const int idx = nn_idx[(b * SS + s) * KK + k];
        lds_idx[t] = idx;

        float d[3];
        #pragma unroll
        for (int j = 0; j < 3; ++j) {
            d[j] = xyz[(b * 3 + j) * NN + idx] - new_xyz[(b * 3 + j) * SS + s];
            lds_gxn[(sl * 3 + j) * 16 + k] = d[j];
        }

        // layer0: 3 -> 8, LN, leaky
        float h0[8];
        {
            float mu = 0.f;
            #pragma unroll
            for (int i = 0; i < 8; ++i) {
                float a = b0[i];
                #pragma unroll
                for (int j = 0; j < 3; ++j) a += d[j] * w0[j * 8 + i];
                h0[i] = a; mu += a;
            }
            mu *= 0.125f;
            float var = 0.f;
            #pragma unroll
            for (int i = 0; i < 8; ++i) { float z = h0[i] - mu; var += z * z; }
            float r = rsqrtf(var * 0.125f + 1e-5f);
            #pragma unroll
            for (int i = 0; i < 8; ++i) h0[i] = leaky02((h0[i] - mu) * r * g0[i] + be0[i]);
        }
        // layer1: 8 -> 8, LN, leaky
        float h1[8];
        {
            float mu = 0.f;
            #pragma unroll
            for (int i = 0; i < 8; ++i) {
                float a = b1[i];
                #pragma unroll
                for (int j = 0; j < 8; ++j) a += h0[j] * w1[j * 8 + i];
                h1[i] = a; mu += a;
            }
            mu *= 0.125f;
            float var = 0.f;
            #pragma unroll
            for (int i = 0; i < 8; ++i) { float z = h1[i] - mu; var += z * z; }
            float r = rsqrtf(var * 0.125f + 1e-5f);
            #pragma unroll
            for (int i = 0; i < 8; ++i) h1[i] = leaky02((h1[i] - mu) * r * g1[i] + be1[i]);
        }
        // layer2: 8 -> 16, LN, leaky
        float h2[16];
        {
            float mu = 0.f;
            #pragma unroll
            for (int i = 0; i < 16; ++i) {
                float a = b2[i];
                #pragma unroll
                for (int j = 0; j < 8; ++j) a += h1[j] * w2[j * 16 + i];
                h2[i] = a; mu += a;
            }
            mu *= (1.0f / 16.0f);
            float var = 0.f;
            #pragma unroll
            for (int i = 0; i < 16; ++i) { float z = h2[i] - mu; var += z * z; }
            float r = rsqrtf(var * (1.0f / 16.0f) + 1e-5f);
            #pragma unroll
            for (int i = 0; i < 16; ++i)
                lds_w[sl * 256 + k * 16 + i] = leaky02((h2[i] - mu) * r * g2[i] + be2[i]);
        }
    }
    __syncthreads();

    // Per-wave output column tile: o = wave*16 + ll
    const int o = wave * 16 + ll;
    const float bl_o  = bl[o];
    const float gl_o  = gl[o];
    const float bel_o = bel[o];

    v8f acc0 = {};
    v8f acc1 = {};

    // ---------- Main loop over 5 channel tiles (67 channels zero-padded to 80) ----------
    for (int ct = 0; ct < 5; ++ct) {
        // Fill feat tile: feat[s][c][k], c global = ct*16 + c_local
        #pragma unroll
        for (int it = 0; it < 16; ++it) {
            int e  = t + 256 * it;          // e == sl*256 + cl*16 + k
            int sl = e >> 8;
            int cl = (e >> 4) & 15;
            int k  = e & 15;
            int cg = ct * 16 + cl;
            float v = 0.f;
            if (cg < 64)      v = points[(b * 64 + cg) * NN + lds_idx[sl * 16 + k]];
            else if (cg < 67) v = lds_gxn[(sl * 3 + (cg - 64)) * 16 + k];
            lds_feat[e] = v;
        }
        __syncthreads();

        // Einsum: per s, agg[16c x 16m] = feat[16c x 16k] * w[16k x 16m]
        // each wave handles 2 s values; K=16 in four 16x16x4 WMMA steps
        #pragma unroll
        for (int sl2 = 0; sl2 < 2; ++sl2) {
            int sl = wave * 2 + sl2;
            v8f c = {};
            #pragma unroll
            for (int kk = 0; kk < 4; ++kk) {
                int kbase = 4 * kk + 2 * lh;      // K = j + 2*half + 4*kk
                v2f a, bv;
                a.x  = lds_feat[sl * 256 + ll * 16 + kbase];
                a.y  = lds_feat[sl * 256 + ll * 16 + kbase + 1];
                bv.x = lds_w[sl * 256 + kbase * 16 + ll];
                bv.y = lds_w[sl * 256 + (kbase + 1) * 16 + ll];
                c = __builtin_amdgcn_wmma_f32_16x16x4_f32(
                        false, a, false, bv, (short)0, c, false, false);
            }
            // C layout: M = r + 8*half (=c), N = ll (=m)
            #pragma unroll
            for (int r = 0; r < 8; ++r)
                lds_agg[sl * 256 + (r + 8 * lh) * 16 + ll] = c[r];
        }
        __syncthreads();

        // GEMM2 accumulate: out[16s x 16o] += agg[16s x 256] @ wl[256 x 16o]
        // two interleaved accumulators to break the WMMA RAW chain
        for (int step = 0; step < 64; step += 2) {
            {
                int kkl = 4 * step + 2 * lh;
                v2f a, bv;
                a.x = lds_agg[ll * 256 + kkl];
                a.y = lds_agg[ll * 256 + kkl + 1];
                int kg = ct * 256 + kkl;
                bv.x = (kg     < KTOT) ? wl[kg * 128 + o]       : 0.f;
                bv.y = (kg + 1 < KTOT) ? wl[(kg + 1) * 128 + o] : 0.f;
                acc0 = __builtin_amdgcn_wmma_f32_16x16x4_f32(
                           false, a, false, bv, (short)0, acc0, false, false);
            }
            {
                int kkl = 4 * (step + 1) + 2 * lh;
                v2f a, bv;
                a.x = lds_agg[ll * 256 + kkl];
                a.y = lds_agg[ll * 256 + kkl + 1];
                int kg = ct * 256 + kkl;
                bv.x = (kg     < KTOT) ? wl[kg * 128 + o]       : 0.f;
                bv.y = (kg + 1 < KTOT) ? wl[(kg + 1) * 128 + o] : 0.f;
                acc1 = __builtin_amdgcn_wmma_f32_16x16x4_f32(
                           false, a, false, bv, (short)0, acc1, false, false);
            }
        }
        __syncthreads();
    }

    // ---------- Epilogue: +bias, LayerNorm over 128 outputs per s, leaky, store ----------
    float x[8], sum1[8], sum2[8];
    #pragma unroll
    for (int r = 0; r < 8; ++r) {
        x[r] = acc0[r] + acc1[r] + bl_o;
        sum1[r] = x[r];
        sum2[r] = x[r] * x[r];
    }
    // butterfly reduce across the 16 lanes of each half (over the o-slice)
    #pragma unroll
    for (int m = 1; m < 16; m <<= 1) {
        #pragma unroll
        for (int r = 0; r < 8; ++r) {
            sum1[r] += __shfl_xor(sum1[r], m, 32);
            sum2[r] += __shfl_xor(sum2[r], m, 32);
        }
    }
    if (ll == 0) {
        #pragma unroll
        for (int r = 0; r < 8; ++r) {
            int sl = r + 8 * lh;
            lds_rsum[sl * 8 + wave] = sum1[r];
            lds_rsq [sl * 8 + wave] = sum2[r];
        }
    }
    __syncthreads();
    if (t < 16) {
        float su = 0.f, sq = 0.f;
        #pragma unroll
        for (int w = 0; w < 8; ++w) { su += lds_rsum[t * 8 + w]; sq += lds_rsq[t * 8 + w]; }
        float mean = su * (1.0f / 128.0f);
        float var  = sq * (1.0f / 128.0f) - mean * mean;
        lds_mean[t] = mean;
        lds_rstd[t] = rsqrtf(var + 1e-5f);
    }
    __syncthreads();
    #pragma unroll
    for (int r = 0; r < 8; ++r) {
        int sl = r + 8 * lh;
        float y = (x[r] - lds_mean[sl]) * lds_rstd[sl] * gl_o + bel_o;
        y = leaky02(y);
        out[(b * DOUT + o) * SS + s0 + sl] = y;   // [B, 128, S]
    }
}

extern "C" void kernel_launch(void* const* d_in, const int* in_sizes, int n_in,
                              void* d_out, int out_size, void* d_ws, size_t ws_size,
                              hipStream_t stream)
{
    (void)in_sizes; (void)n_in; (void)out_size; (void)d_ws; (void)ws_size;
    const float* xyz     = (const float*)d_in[0];
    const float* points  = (const float*)d_in[1];
    const float* new_xyz = (const float*)d_in[2];
    const int*   nn_idx  = (const int*)  d_in[3];
    const float* w0  = (const float*)d_in[4];
    const float* b0  = (const float*)d_in[5];
    const float* g0  = (const float*)d_in[6];
    const float* be0 = (const float*)d_in[7];
    const float* w1  = (const float*)d_in[8];
    const float* b1  = (const float*)d_in[9];
    const float* g1  = (const float*)d_in[10];
    const float* be1 = (const float*)d_in[11];
    const float* w2  = (const float*)d_in[12];
    const float* b2  = (const float*)d_in[13];
    const float* g2  = (const float*)d_in[14];
    const float* be2 = (const float*)d_in[15];
    const float* wl  = (const float*)d_in[16];
    const float* bl  = (const float*)d_in[17];
    const float* gl  = (const float*)d_in[18];
    const float* bel = (const float*)d_in[19];
    float* out = (float*)d_out;

    dim3 grid(BB * (SS / 16));   // 4096 blocks: one per (batch, 16-s tile)
    dim3 block(256);             // 8 wave32
    pointconv_fused<<<grid, block, 0, stream>>>(
        xyz, points, new_xyz, nn_idx,
        w0, b0, g0, be0, w1, b1, g1, be1, w2, b2, g2, be2,
        wl, bl, gl, bel, out);
}